// shared_encoder_28303834481477
// MI455X (gfx1250) — compile-verified
//
#include <hip/hip_runtime.h>

typedef __attribute__((ext_vector_type(2))) float v2f;
typedef __attribute__((ext_vector_type(4))) float v4f;
typedef __attribute__((ext_vector_type(8))) float v8f;

#define FEATS 64
#define LDS_STRIDE 66   // even (keeps b64 alignment), !=64 (spreads LDS banks)

// ---------------------------------------------------------------------------
// init: zero the output accumulator, set deg = 1.0 (self-loop contribution)
// ---------------------------------------------------------------------------
__global__ void gcn_init(float* __restrict__ out, float* __restrict__ deg, int n) {
  int t = blockIdx.x * blockDim.x + threadIdx.x;
  int total4 = n * (FEATS / 4);
  if (t < total4) {
    v4f z = {0.f, 0.f, 0.f, 0.f};
    ((v4f*)out)[t] = z;
  }
  if (t < n) deg[t] = 1.0f;
}

// ---------------------------------------------------------------------------
// degree accumulation over dst indices (edge_index row 1)
// ---------------------------------------------------------------------------
__global__ void gcn_deg(const int* __restrict__ ei, float* __restrict__ deg, int e) {
  int t = blockIdx.x * blockDim.x + threadIdx.x;
  if (t < e) {
    int d = ei[e + t];
    atomicAdd(&deg[d], 1.0f);
  }
}

// ---------------------------------------------------------------------------
// deg -> deg^{-1/2} in place (deg >= 1 always, self loops included)
// ---------------------------------------------------------------------------
__global__ void gcn_rsqrt(float* __restrict__ deg, int n) {
  int t = blockIdx.x * blockDim.x + threadIdx.x;
  if (t < n) deg[t] = rsqrtf(deg[t]);
}

// ---------------------------------------------------------------------------
// h = x @ W via V_WMMA_F32_16X16X4_F32 (fp32 matrix pipe, wave32)
// block = 128 threads = 4 waves; each wave computes a 16x64 strip of h.
// W staged TRANSPOSED in LDS: wl[col*66 + k] so a B fragment
// {W[kr][col], W[kr+1][col]} is one contiguous 8-byte ds_load_b64.
//
// A fragment (16x4):  lane<16: {K=kr,kr+1}, kr=k;   lane>=16: kr=k+2. Row M=lane&15.
// Out-of-range A rows are CLAMPED (not masked): D row M depends only on A row M,
// and invalid rows are never stored, so garbage is harmless -> unconditional loads.
// C/D (16x16): elem v -> row M = v + 8*(lane>>4), col N = lane&15
// ---------------------------------------------------------------------------
__global__ void __launch_bounds__(128)
gcn_gemm_wmma(const float* __restrict__ x, const float* __restrict__ w,
              float* __restrict__ h, int n) {
  __shared__ float wl[FEATS * LDS_STRIDE];
  for (int i = threadIdx.x; i < FEATS * FEATS; i += 128) {
    int r = i >> 6;      // k
    int c = i & 63;      // col
    wl[c * LDS_STRIDE + r] = w[i];
  }
  __syncthreads();

  const int wave = threadIdx.x >> 5;
  const int lane = threadIdx.x & 31;
  const int half = lane >> 4;
  const int lc   = lane & 15;

  const int row0 = (blockIdx.x * 4 + wave) * 16;
  if (row0 >= n) return;                       // wave-uniform exit
  const bool full = (row0 + 16) <= n;          // wave-uniform

  int arow = row0 + lc;
  if (arow >= n) arow = n - 1;                 // clamp: safe addr, value unused
  const float* xr = x + (size_t)arow * FEATS + 2 * half;

  const float* bp0 = wl + (size_t)( 0 + lc) * LDS_STRIDE + 2 * half;
  const float* bp1 = wl + (size_t)(16 + lc) * LDS_STRIDE + 2 * half;
  const float* bp2 = wl + (size_t)(32 + lc) * LDS_STRIDE + 2 * half;
  const float* bp3 = wl + (size_t)(48 + lc) * LDS_STRIDE + 2 * half;

  v8f a0 = {}, a1 = {}, a2 = {}, a3 = {};

#pragma unroll
  for (int k = 0; k < FEATS; k += 4) {
    v2f A  = *(const v2f*)(xr  + k);
    v2f B0 = *(const v2f*)(bp0 + k);
    v2f B1 = *(const v2f*)(bp1 + k);
    v2f B2 = *(const v2f*)(bp2 + k);
    v2f B3 = *(const v2f*)(bp3 + k);
    a0 = __builtin_amdgcn_wmma_f32_16x16x4_f32(false, A, false, B0, (short)0, a0, false, false);
    a1 = __builtin_amdgcn_wmma_f32_16x16x4_f32(false, A, false, B1, (short)0, a1, false, false);
    a2 = __builtin_amdgcn_wmma_f32_16x16x4_f32(false, A, false, B2, (short)0, a2, false, false);
    a3 = __builtin_amdgcn_wmma_f32_16x16x4_f32(false, A, false, B3, (short)0, a3, false, false);
  }

  if (full) {
#pragma unroll
    for (int v = 0; v < 8; ++v) {
      int r = row0 + v + 8 * half;
      float* hr = h + (size_t)r * FEATS + lc;
      hr[0]  = a0[v];
      hr[16] = a1[v];
      hr[32] = a2[v];
      hr[48] = a3[v];
    }
  } else {
#pragma unroll
    for (int v = 0; v < 8; ++v) {
      int r = row0 + v + 8 * half;
      if (r < n) {
        float* hr = h + (size_t)r * FEATS + lc;
        hr[0]  = a0[v];
        hr[16] = a1[v];
        hr[32] = a2[v];
        hr[48] = a3[v];
      }
    }
  }
}

// ---------------------------------------------------------------------------
// scatter: out[dst] += h[src] * (dinv[src]*dinv[dst])
// 16 threads per edge, 4 strided feature atomics each (coalesced in groups)
// ---------------------------------------------------------------------------
__global__ void gcn_scatter(const float* __restrict__ h, const int* __restrict__ ei,
                            const float* __restrict__ dinv, float* __restrict__ out,
                            int e) {
  int t = blockIdx.x * blockDim.x + threadIdx.x;
  int edge = t >> 4;
  int fg   = t & 15;
  if (edge >= e) return;
  int s = ei[edge];        // edge_index[0][edge]
  int d = ei[e + edge];    // edge_index[1][edge]
  float norm = dinv[s] * dinv[d];
  const float* hs = h + (size_t)s * FEATS;
  float* od = out + (size_t)d * FEATS;
#pragma unroll
  for (int j = 0; j < 4; ++j) {
    int f = fg + 16 * j;
    atomicAdd(&od[f], hs[f] * norm);
  }
}

// ---------------------------------------------------------------------------
// finalize: out = relu(out + h * dinv^2 (self-loop) + bias)
// ---------------------------------------------------------------------------
__global__ void gcn_finalize(float* __restrict__ out, const float* __restrict__ h,
                             const float* __restrict__ dinv,
                             const float* __restrict__ bias, int n) {
  int t = blockIdx.x * blockDim.x + threadIdx.x;
  int node = t >> 4;
  int g = (t & 15) * 4;
  if (node >= n) return;
  float di = dinv[node];
  float sl = di * di;  // 1/deg
  size_t base = (size_t)node * FEATS + g;
  v4f o  = *(const v4f*)(out + base);
  v4f hh = *(const v4f*)(h + base);
  v4f b  = *(const v4f*)(bias + g);
#pragma unroll
  for (int j = 0; j < 4; ++j) {
    float r = o[j] + hh[j] * sl + b[j];
    o[j] = r > 0.f ? r : 0.f;
  }
  *(v4f*)(out + base) = o;
}

// ---------------------------------------------------------------------------
extern "C" void kernel_launch(void* const* d_in, const int* in_sizes, int n_in,
                              void* d_out, int out_size, void* d_ws, size_t ws_size,
                              hipStream_t stream) {
  const float* x    = (const float*)d_in[0];
  const int*   ei   = (const int*)d_in[1];   // [2, E] flattened
  const float* w    = (const float*)d_in[2]; // [64, 64]
  const float* bias = (const float*)d_in[3]; // [64]
  float* out = (float*)d_out;

  const int N = in_sizes[0] / FEATS;
  const int E = in_sizes[1] / 2;

  float* h   = (float*)d_ws;                 // N*64 floats (25.6 MB)
  float* deg = h + (size_t)N * FEATS;        // N floats; becomes dinv in place

  const int B = 256;

  // 1) init out=0, deg=1 (self-loop)
  {
    int threads = N * (FEATS / 4);
    gcn_init<<<(threads + B - 1) / B, B, 0, stream>>>(out, deg, N);
  }
  // 2) degree atomics over edges
  gcn_deg<<<(E + B - 1) / B, B, 0, stream>>>(ei, deg, E);
  // 3) deg -> deg^{-1/2}
  gcn_rsqrt<<<(N + B - 1) / B, B, 0, stream>>>(deg, N);
  // 4) h = x @ W  (WMMA fp32)
  {
    int blocks = (N + 63) / 64;  // 4 waves x 16 rows per block
    gcn_gemm_wmma<<<blocks, 128, 0, stream>>>(x, w, h, N);
  }
  // 5) edge scatter with symmetric normalization
  {
    long long threads = (long long)E * 16;
    gcn_scatter<<<(int)((threads + B - 1) / B), B, 0, stream>>>(h, ei, deg, out, E);
  }
  // 6) self-loop + bias + relu
  {
    int threads = N * 16;
    gcn_finalize<<<(threads + B - 1) / B, B, 0, stream>>>(out, h, deg, bias, N);
  }
}